// GraphAttentionMLP2_33268816675090
// MI455X (gfx1250) — compile-verified
//
#include <hip/hip_runtime.h>
#include <hip/hip_bf16.h>

typedef __attribute__((ext_vector_type(2))) float v2f;
typedef __attribute__((ext_vector_type(8))) float v8f;

#define WMMA_F32(a, b, c) \
  __builtin_amdgcn_wmma_f32_16x16x4_f32(false, (a), false, (b), (short)0, (c), false, false)

__device__ __forceinline__ float sigmoidf_(float x) { return 1.0f / (1.0f + __expf(-x)); }
__device__ __forceinline__ float siluf_(float x) { return x * sigmoidf_(x); }
__device__ __forceinline__ float slreluf_(float x) {
  // smooth leaky relu, a = 0.2: 0.6*x + 0.4*x*(2*sigmoid(x)-1)
  return 0.6f * x + 0.4f * x * (2.0f * sigmoidf_(x) - 1.0f);
}

__device__ __forceinline__ void atomicMaxFloat(float* addr, float val) {
  unsigned int* ai = (unsigned int*)addr;
  unsigned int old = *ai;
  while (true) {
    float cur = __uint_as_float(old);
    if (cur >= val) break;
    unsigned int assumed = old;
    old = atomicCAS(ai, assumed, __float_as_uint(val));
    if (old == assumed) break;
  }
}

// ---------------------------------------------------------------------------
// Init: m = -inf, z = 0, attn = 0
// ---------------------------------------------------------------------------
__global__ void init_kernel(float* __restrict__ ws_m, float* __restrict__ ws_z,
                            float* __restrict__ ws_attn, int N) {
  int t = blockIdx.x * blockDim.x + threadIdx.x;
  if (t < N * 96) ws_attn[t] = 0.0f;
  if (t < N * 8) {
    ws_m[t] = -1e30f;
    ws_z[t] = 0.0f;
  }
}

// ---------------------------------------------------------------------------
// Edge kernel: one wave per 16-edge tile; 4 waves (128 threads) per block.
// Chains 5 WMMA f32 GEMMs with fused activations.
// ---------------------------------------------------------------------------
__global__ __launch_bounds__(128) void edge_kernel(
    const float* __restrict__ message,      // [E,96]
    const float* __restrict__ edge_attr,    // [E,1]
    const float* __restrict__ edge_scalars, // [E,32]
    const float* __restrict__ W_rad1,       // [32,32]
    const float* __restrict__ b_rad1,       // [32]
    const float* __restrict__ W_rad2,       // [32,96]
    const float* __restrict__ b_rad2,       // [96]
    const float* __restrict__ W_alpha,      // [96,96]
    const float* __restrict__ W_lin,        // [96,96]
    const float* __restrict__ w_dtp2,       // [96]
    const float* __restrict__ W_lin2,       // [96,96]
    const float* __restrict__ alpha_dot,    // [8,12]
    const int* __restrict__ edge_dst,       // [E]
    float* __restrict__ ws_value,           // [E,96]
    float* __restrict__ ws_logits,          // [E,8]
    float* __restrict__ ws_m,               // [N,8]
    int E) {
  __shared__ float lds_hid[4][16][33];   // radial hidden
  __shared__ float lds_msg[4][16][97];   // msg tile (A operand)
  __shared__ float lds_val[4][16][97];   // gated value tile (A operand)
  __shared__ float lds_logit[4][16][8];  // per-edge per-head logits
  __shared__ float lds_ea[4][16];        // edge_attr tile

  const int wv = threadIdx.x >> 5;
  const int lane = threadIdx.x & 31;
  const int e0 = (blockIdx.x * 4 + wv) * 16;
  const int half = lane >> 4;  // 0: lanes 0-15, 1: lanes 16-31
  const int mrow = lane & 15;  // row for A frags / column for B frags
  const int koff = half * 2;   // K sub-offset inside the 4-wide K step

  int erow = e0 + mrow;
  if (erow >= E) erow = E - 1;

  if (lane < 16) {
    int e = e0 + lane;
    if (e >= E) e = E - 1;
    lds_ea[wv][lane] = edge_attr[e];
#pragma unroll
    for (int h = 0; h < 8; ++h) lds_logit[wv][lane][h] = 0.0f;
  }
  __syncthreads();

  // ---- GEMM1: hid = silu(edge_scalars @ W_rad1 + b_rad1)  [16,32] ----
#pragma unroll
  for (int j = 0; j < 2; ++j) {
    v8f c = {};
#pragma unroll
    for (int kk = 0; kk < 8; ++kk) {
      const int k0 = kk * 4;
      v2f a, b;
      a.x = edge_scalars[(size_t)erow * 32 + k0 + koff];
      a.y = edge_scalars[(size_t)erow * 32 + k0 + koff + 1];
      const int n = j * 16 + mrow;
      b.x = W_rad1[(k0 + koff) * 32 + n];
      b.y = W_rad1[(k0 + koff + 1) * 32 + n];
      c = WMMA_F32(a, b, c);
    }
    const int n = j * 16 + mrow;
    const float bias = b_rad1[n];
#pragma unroll
    for (int r = 0; r < 8; ++r) {
      const int m = r + half * 8;
      lds_hid[wv][m][n] = siluf_(c[r] + bias);
    }
  }
  __syncthreads();

  // ---- GEMM2: weight = hid @ W_rad2 + b_rad2; msg = message*ea*weight ----
#pragma unroll
  for (int j = 0; j < 6; ++j) {
    v8f c = {};
#pragma unroll
    for (int kk = 0; kk < 8; ++kk) {
      const int k0 = kk * 4;
      v2f a, b;
      a.x = lds_hid[wv][mrow][k0 + koff];
      a.y = lds_hid[wv][mrow][k0 + koff + 1];
      const int n = j * 16 + mrow;
      b.x = W_rad2[(k0 + koff) * 96 + n];
      b.y = W_rad2[(k0 + koff + 1) * 96 + n];
      c = WMMA_F32(a, b, c);
    }
    const int n = j * 16 + mrow;
    const float bias = b_rad2[n];
#pragma unroll
    for (int r = 0; r < 8; ++r) {
      const int m = r + half * 8;
      int e = e0 + m;
      if (e >= E) e = E - 1;
      const float mval = message[(size_t)e * 96 + n];
      lds_msg[wv][m][n] = mval * lds_ea[wv][m] * (c[r] + bias);
    }
  }
  __syncthreads();

  // ---- GEMM3: log_alpha = msg @ W_alpha; smooth-leaky + head reduction ----
#pragma unroll
  for (int j = 0; j < 6; ++j) {
    v8f c = {};
#pragma unroll
    for (int kk = 0; kk < 24; ++kk) {
      const int k0 = kk * 4;
      v2f a, b;
      a.x = lds_msg[wv][mrow][k0 + koff];
      a.y = lds_msg[wv][mrow][k0 + koff + 1];
      const int n = j * 16 + mrow;
      b.x = W_alpha[(k0 + koff) * 96 + n];
      b.y = W_alpha[(k0 + koff + 1) * 96 + n];
      c = WMMA_F32(a, b, c);
    }
    const int n = j * 16 + mrow;
    const int h = n / 12;
    const float ad = alpha_dot[n];  // alpha_dot[h][n%12] flattened
#pragma unroll
    for (int r = 0; r < 8; ++r) {
      const int m = r + half * 8;
      atomicAdd(&lds_logit[wv][m][h], slreluf_(c[r]) * ad);
    }
  }

  // ---- GEMM4: value = silu(msg @ W_lin) * ea * w_dtp2 -> lds_val ----
#pragma unroll
  for (int j = 0; j < 6; ++j) {
    v8f c = {};
#pragma unroll
    for (int kk = 0; kk < 24; ++kk) {
      const int k0 = kk * 4;
      v2f a, b;
      a.x = lds_msg[wv][mrow][k0 + koff];
      a.y = lds_msg[wv][mrow][k0 + koff + 1];
      const int n = j * 16 + mrow;
      b.x = W_lin[(k0 + koff) * 96 + n];
      b.y = W_lin[(k0 + koff + 1) * 96 + n];
      c = WMMA_F32(a, b, c);
    }
    const int n = j * 16 + mrow;
    const float wd = w_dtp2[n];
#pragma unroll
    for (int r = 0; r < 8; ++r) {
      const int m = r + half * 8;
      lds_val[wv][m][n] = siluf_(c[r]) * lds_ea[wv][m] * wd;
    }
  }
  __syncthreads();

  // ---- GEMM5: value2 = lds_val @ W_lin2 -> ws_value ----
#pragma unroll
  for (int j = 0; j < 6; ++j) {
    v8f c = {};
#pragma unroll
    for (int kk = 0; kk < 24; ++kk) {
      const int k0 = kk * 4;
      v2f a, b;
      a.x = lds_val[wv][mrow][k0 + koff];
      a.y = lds_val[wv][mrow][k0 + koff + 1];
      const int n = j * 16 + mrow;
      b.x = W_lin2[(k0 + koff) * 96 + n];
      b.y = W_lin2[(k0 + koff + 1) * 96 + n];
      c = WMMA_F32(a, b, c);
    }
    const int n = j * 16 + mrow;
#pragma unroll
    for (int r = 0; r < 8; ++r) {
      const int m = r + half * 8;
      const int e = e0 + m;
      if (e < E) ws_value[(size_t)e * 96 + n] = c[r];
    }
  }
  __syncthreads();  // make lds_logit atomics visible

  // ---- logits out + segment max (atomic) ----
  if (lane < 16) {
    const int e = e0 + lane;
    if (e < E) {
      const int dst = edge_dst[e];
#pragma unroll
      for (int h = 0; h < 8; ++h) {
        const float lg = lds_logit[wv][lane][h];
        ws_logits[(size_t)e * 8 + h] = lg;
        atomicMaxFloat(&ws_m[(size_t)dst * 8 + h], lg);
      }
    }
  }
}

// ---------------------------------------------------------------------------
// Scatter: p = exp(logit - m[dst]); attn[dst] += value2*p; z[dst] += p
// ---------------------------------------------------------------------------
__global__ void scatter_kernel(const float* __restrict__ ws_value,
                               const float* __restrict__ ws_logits,
                               const float* __restrict__ ws_m,
                               const int* __restrict__ edge_dst,
                               float* __restrict__ ws_attn,
                               float* __restrict__ ws_z, int E) {
  const long long t = (long long)blockIdx.x * blockDim.x + threadIdx.x;
  if (t >= (long long)E * 96) return;
  const int e = (int)(t / 96);
  const int c = (int)(t % 96);
  const int h = c / 12;
  const int dst = edge_dst[e];
  const float p = __expf(ws_logits[(size_t)e * 8 + h] - ws_m[(size_t)dst * 8 + h]);
  atomicAdd(&ws_attn[(size_t)dst * 96 + c], ws_value[t] * p);
  if (c % 12 == 0) atomicAdd(&ws_z[(size_t)dst * 8 + h], p);
}

// ---------------------------------------------------------------------------
// Node kernel: out = (attn / z) @ W_proj, WMMA over 16-node tiles
// ---------------------------------------------------------------------------
__global__ __launch_bounds__(128) void node_kernel(const float* __restrict__ ws_attn,
                                                   const float* __restrict__ ws_z,
                                                   const float* __restrict__ W_proj,
                                                   float* __restrict__ out, int N) {
  __shared__ float lds_a[4][16][97];
  const int wv = threadIdx.x >> 5;
  const int lane = threadIdx.x & 31;
  const int n0 = (blockIdx.x * 4 + wv) * 16;

  for (int i = lane; i < 16 * 96; i += 32) {
    const int m = i / 96;
    const int c = i % 96;
    int nd = n0 + m;
    if (nd >= N) nd = N - 1;
    const float z = ws_z[(size_t)nd * 8 + c / 12];
    const float a = ws_attn[(size_t)nd * 96 + c];
    lds_a[wv][m][c] = (z > 0.0f) ? (a / z) : 0.0f;
  }
  __syncthreads();

  const int half = lane >> 4;
  const int mrow = lane & 15;
  const int koff = half * 2;
#pragma unroll
  for (int j = 0; j < 6; ++j) {
    v8f c = {};
#pragma unroll
    for (int kk = 0; kk < 24; ++kk) {
      const int k0 = kk * 4;
      v2f a, b;
      a.x = lds_a[wv][mrow][k0 + koff];
      a.y = lds_a[wv][mrow][k0 + koff + 1];
      const int n = j * 16 + mrow;
      b.x = W_proj[(k0 + koff) * 96 + n];
      b.y = W_proj[(k0 + koff + 1) * 96 + n];
      c = WMMA_F32(a, b, c);
    }
    const int ncol = j * 16 + mrow;
#pragma unroll
    for (int r = 0; r < 8; ++r) {
      const int m = r + half * 8;
      const int nd = n0 + m;
      if (nd < N) out[(size_t)nd * 96 + ncol] = c[r];
    }
  }
}

// ---------------------------------------------------------------------------
extern "C" void kernel_launch(void* const* d_in, const int* in_sizes, int n_in,
                              void* d_out, int out_size, void* d_ws, size_t ws_size,
                              hipStream_t stream) {
  const float* message      = (const float*)d_in[0];
  const float* edge_attr    = (const float*)d_in[1];
  const float* edge_scalars = (const float*)d_in[2];
  const float* W_rad1       = (const float*)d_in[3];
  const float* b_rad1       = (const float*)d_in[4];
  const float* W_rad2       = (const float*)d_in[5];
  const float* b_rad2       = (const float*)d_in[6];
  const float* W_alpha      = (const float*)d_in[7];
  const float* W_lin        = (const float*)d_in[8];
  const float* w_dtp2       = (const float*)d_in[9];
  const float* W_lin2       = (const float*)d_in[10];
  const float* alpha_dot    = (const float*)d_in[11];
  const int*   edge_dst     = (const int*)d_in[12];
  const float* W_proj       = (const float*)d_in[12 - 2]; // placeholder fixed below

  const int E = in_sizes[1];    // edge_attr is [E,1]
  const int N = out_size / 96;  // output is [N,96]

  // NOTE: correct input order per setup_inputs():
  // 0 message, 1 edge_attr, 2 edge_scalars, 3 W_rad1, 4 b_rad1, 5 W_rad2,
  // 6 b_rad2, 7 W_alpha, 8 W_lin, 9 w_dtp2, 10 W_lin2, 11 alpha_dot,
  // 12 W_proj, 13 edge_dst, 14 n_nodes_dst
  const float* W_proj_real  = (const float*)d_in[12];
  const int*   edge_dst_real = (const int*)d_in[13];
  (void)edge_dst; (void)W_proj;

  float* ws = (float*)d_ws;
  float* ws_value  = ws;                              // E*96
  float* ws_logits = ws_value + (size_t)E * 96;       // E*8
  float* ws_m      = ws_logits + (size_t)E * 8;       // N*8
  float* ws_z      = ws_m + (size_t)N * 8;            // N*8
  float* ws_attn   = ws_z + (size_t)N * 8;            // N*96

  {
    int total = N * 96;
    init_kernel<<<(total + 255) / 256, 256, 0, stream>>>(ws_m, ws_z, ws_attn, N);
  }
  {
    int blocks = (E + 63) / 64;  // 4 waves x 16 edges per block
    edge_kernel<<<blocks, 128, 0, stream>>>(
        message, edge_attr, edge_scalars, W_rad1, b_rad1, W_rad2, b_rad2,
        W_alpha, W_lin, w_dtp2, W_lin2, alpha_dot, edge_dst_real,
        ws_value, ws_logits, ws_m, E);
  }
  {
    long long total = (long long)E * 96;
    int blocks = (int)((total + 255) / 256);
    scatter_kernel<<<blocks, 256, 0, stream>>>(ws_value, ws_logits, ws_m,
                                               edge_dst_real, ws_attn, ws_z, E);
  }
  {
    int blocks = (N + 63) / 64;
    node_kernel<<<blocks, 128, 0, stream>>>(ws_attn, ws_z, W_proj_real,
                                            (float*)d_out, N);
  }
}